// RecurrentGConvGRU_40037685133529
// MI455X (gfx1250) — compile-verified
//
#include <hip/hip_runtime.h>

// GConvGRU (Chebyshev K=3, H0=0) for MI455X / gfx1250, wave32 + WMMA fp32.
//
// Reduction (H0 == 0):
//   cheb(H0,*)        == bias only, R path irrelevant (H0*R == 0)
//   Z  = sigmoid(Tx0@Wxz0 + Tx1@Wxz1 + Tx2@Wxz2 + bxz + bhz)
//   Ht = tanh   (Tx0@Wxh0 + Tx1@Wxh1 + Tx2@Wxh2 + bxh + bhh)
//   out = relu((1-Z)*Ht) @ Wlin + blin
// with Tx0 = x, Tx1 = scatter(w*x[src]->dst), Tx2 = 2*scatter(w*Tx1[src]) - x,
// w_e = -ew * dinv[src] * dinv[dst], deg from src.

typedef __attribute__((ext_vector_type(2))) float v2f;
typedef __attribute__((ext_vector_type(8))) float v8f;

#define NN 50000
#define NE 800000
#define FD 64
#define NTILES (NN / 16)  // 3125, exact

#define LOG2E 1.4426950408889634f

// V_WMMA_F32_16X16X4_F32: A 16x4 f32 (2 VGPR), B 4x16 f32 (2 VGPR), C/D 16x16 f32 (8 VGPR)
__device__ __forceinline__ v8f wmma4(v2f a, v2f b, v8f c) {
  return __builtin_amdgcn_wmma_f32_16x16x4_f32(
      /*neg_a=*/false, a, /*neg_b=*/false, b,
      /*c_mod=*/(short)0, c, /*reuse_a=*/false, /*reuse_b=*/false);
}

// Branch-free gate nonlinearities on the hardware transcendental pipe
// (v_exp_f32 / v_rcp_f32 / v_tanh_f32) instead of branchy libm expansions.
__device__ __forceinline__ float fast_sigmoid(float x) {
  float t = __builtin_amdgcn_exp2f(-x * LOG2E);
  return __builtin_amdgcn_rcpf(1.0f + t);
}

__device__ __forceinline__ float fast_tanh(float x) {
#if __has_builtin(__builtin_amdgcn_tanhf)
  return __builtin_amdgcn_tanhf(x);  // native v_tanh_f32 on gfx1250
#else
  // tanh(x) = 1 - 2/(exp(2x)+1); exp overflow -> rcp(inf)=0 -> 1 (correct tail)
  float e = __builtin_amdgcn_exp2f(x * (2.0f * LOG2E));
  return 1.0f - 2.0f * __builtin_amdgcn_rcpf(e + 1.0f);
#endif
}

__global__ __launch_bounds__(256) void zero_kernel(float* __restrict__ p, int n) {
  int i = blockIdx.x * blockDim.x + threadIdx.x;
  int stride = gridDim.x * blockDim.x;
  for (; i < n; i += stride) p[i] = 0.0f;
}

__global__ __launch_bounds__(256) void deg_kernel(const float* __restrict__ ew,
                                                  const int* __restrict__ src,
                                                  float* __restrict__ deg) {
  int e = blockIdx.x * 256 + threadIdx.x;
  if (e < NE) unsafeAtomicAdd(deg + src[e], ew[e]);
}

__global__ __launch_bounds__(256) void dinv_kernel(const float* __restrict__ deg,
                                                   float* __restrict__ dinv) {
  int i = blockIdx.x * 256 + threadIdx.x;
  if (i < NN) {
    float d = deg[i];
    dinv[i] = (d > 0.0f) ? rsqrtf(d) : 0.0f;
  }
}

__global__ __launch_bounds__(256) void lapw_kernel(const float* __restrict__ ew,
                                                   const int* __restrict__ src,
                                                   const int* __restrict__ dst,
                                                   const float* __restrict__ dinv,
                                                   float* __restrict__ wl) {
  int e = blockIdx.x * 256 + threadIdx.x;
  if (e < NE) wl[e] = -ew[e] * dinv[src[e]] * dinv[dst[e]];
}

// acc[dst] += w_e * in[src]   (16 threads per edge, float4 per thread)
__global__ __launch_bounds__(256) void spmm_kernel(const float* __restrict__ wl,
                                                   const int* __restrict__ src,
                                                   const int* __restrict__ dst,
                                                   const float* __restrict__ in,
                                                   float* __restrict__ acc) {
  int gid = blockIdx.x * 256 + threadIdx.x;
  int e = gid >> 4;
  if (e >= NE) return;
  int f = (gid & 15) << 2;
  float we = wl[e];
  int s = src[e];
  int d = dst[e];
  const float4 xv = *(const float4*)(in + (size_t)s * FD + f);
  float* o = acc + (size_t)d * FD + f;
  unsafeAtomicAdd(o + 0, we * xv.x);
  unsafeAtomicAdd(o + 1, we * xv.y);
  unsafeAtomicAdd(o + 2, we * xv.z);
  unsafeAtomicAdd(o + 3, we * xv.w);
}

// Fused dense stage. One wave per 16-node tile, full 64 output columns.
// Tx2 fragment computed in registers as 2*t2 - x.
__global__ __launch_bounds__(256) void dense_kernel(
    const float* __restrict__ x, const float* __restrict__ tx1,
    const float* __restrict__ t2, const float* __restrict__ Wxz,
    const float* __restrict__ bxz, const float* __restrict__ bhz,
    const float* __restrict__ Wxh, const float* __restrict__ bxh,
    const float* __restrict__ bhh, const float* __restrict__ Wlin,
    const float* __restrict__ blin, float* __restrict__ out) {
  __shared__ float ldsH[8][16 * 66];  // stride 66: 8B aligned, bank-friendly
  const int wave = threadIdx.x >> 5;
  const int lane = threadIdx.x & 31;
  const int tile = blockIdx.x * 8 + wave;
  if (tile >= NTILES) return;  // wave-uniform: EXEC stays all-ones for WMMA
  const int m0 = tile << 4;
  const int rl = lane & 15;            // A row / B-D column within tile
  const int hh2 = (lane >> 4) << 1;    // K offset: 0 (lanes 0-15) or 2 (16-31)
  const int rowoff = (lane >> 4) << 3; // C/D row offset: 0 or 8

  const float* xr = x + (size_t)(m0 + rl) * FD;
  const float* t1r = tx1 + (size_t)(m0 + rl) * FD;
  const float* t2r = t2 + (size_t)(m0 + rl) * FD;

  const v8f zero8 = {0.f, 0.f, 0.f, 0.f, 0.f, 0.f, 0.f, 0.f};
  v8f accz[4] = {zero8, zero8, zero8, zero8};
  v8f acch[4] = {zero8, zero8, zero8, zero8};

  for (int k0 = 0; k0 < FD; k0 += 4) {
    const int ka = k0 + hh2;
    v2f a0 = *(const v2f*)(xr + ka);                 // Tx0 fragment
    v2f a1 = *(const v2f*)(t1r + ka);                // Tx1 fragment
    v2f a2 = 2.0f * (*(const v2f*)(t2r + ka)) - a0;  // Tx2 fragment
    const float* wz = Wxz + ka * FD;  // W[k] stride = 64*64 = 4096
    const float* wh = Wxh + ka * FD;
#pragma unroll
    for (int ct = 0; ct < 4; ++ct) {
      const int n = (ct << 4) + rl;
      v2f b;
      b.x = wz[n];        b.y = wz[FD + n];        accz[ct] = wmma4(a0, b, accz[ct]);
      b.x = wz[4096 + n]; b.y = wz[4096 + FD + n]; accz[ct] = wmma4(a1, b, accz[ct]);
      b.x = wz[8192 + n]; b.y = wz[8192 + FD + n]; accz[ct] = wmma4(a2, b, accz[ct]);
      b.x = wh[n];        b.y = wh[FD + n];        acch[ct] = wmma4(a0, b, acch[ct]);
      b.x = wh[4096 + n]; b.y = wh[4096 + FD + n]; acch[ct] = wmma4(a1, b, acch[ct]);
      b.x = wh[8192 + n]; b.y = wh[8192 + FD + n]; acch[ct] = wmma4(a2, b, acch[ct]);
    }
  }

  // Gate nonlinearity in D-layout, stage H = relu((1-Z)*tanh) through LDS
  float* Hl = ldsH[wave];
#pragma unroll
  for (int ct = 0; ct < 4; ++ct) {
    const int n = (ct << 4) + rl;
    const float bz = bxz[n] + bhz[n];
    const float bh = bxh[n] + bhh[n];
#pragma unroll
    for (int v = 0; v < 8; ++v) {
      const float zf = fast_sigmoid(accz[ct][v] + bz);
      const float ht = fast_tanh(acch[ct][v] + bh);
      float Hv = (1.0f - zf) * ht;
      Hv = Hv > 0.0f ? Hv : 0.0f;
      Hl[(rowoff + v) * 66 + n] = Hv;  // row (v+8h), col n
    }
  }
  // same-wave producer/consumer: wait LDS stores before re-reading as A frags
  asm volatile("s_wait_dscnt 0x0" ::: "memory");

  // out = H @ Wlin + blin
  v8f acco[4] = {zero8, zero8, zero8, zero8};
  for (int k0 = 0; k0 < FD; k0 += 4) {
    const int ka = k0 + hh2;
    v2f a = *(const v2f*)(&Hl[rl * 66 + ka]);
    const float* wlp = Wlin + ka * FD;
#pragma unroll
    for (int ct = 0; ct < 4; ++ct) {
      const int n = (ct << 4) + rl;
      v2f b;
      b.x = wlp[n];
      b.y = wlp[FD + n];
      acco[ct] = wmma4(a, b, acco[ct]);
    }
  }
  float* orow = out + (size_t)m0 * FD;
#pragma unroll
  for (int ct = 0; ct < 4; ++ct) {
    const int n = (ct << 4) + rl;
    const float bo = blin[n];
#pragma unroll
    for (int v = 0; v < 8; ++v) {
      orow[(size_t)(rowoff + v) * FD + n] = acco[ct][v] + bo;
    }
  }
}

extern "C" void kernel_launch(void* const* d_in, const int* in_sizes, int n_in,
                              void* d_out, int out_size, void* d_ws, size_t ws_size,
                              hipStream_t stream) {
  (void)in_sizes; (void)n_in; (void)out_size; (void)ws_size;
  const float* x    = (const float*)d_in[0];
  const int*   eidx = (const int*)d_in[1];     // (2,E): [0:E]=src, [E:2E]=dst
  const float* ew   = (const float*)d_in[2];
  const float* Wxz  = (const float*)d_in[3];
  const float* bxz  = (const float*)d_in[4];
  const float* bhz  = (const float*)d_in[6];   // W_hz unused (H0==0)
  // d_in[7..10] (R path) provably unused: H0*R == 0
  const float* Wxh  = (const float*)d_in[11];
  const float* bxh  = (const float*)d_in[12];
  const float* bhh  = (const float*)d_in[14];  // W_hh unused (H0==0)
  const float* Wlin = (const float*)d_in[15];
  const float* blin = (const float*)d_in[16];
  float* out = (float*)d_out;

  const int* src = eidx;
  const int* dst = eidx + NE;

  // Workspace layout (floats): deg[NN] | dinv[NN] | wlap[NE] | Tx1[NN*64] | T2[NN*64]
  float* ws   = (float*)d_ws;
  float* deg  = ws;
  float* dinv = ws + NN;
  float* wlap = ws + 2 * NN;
  float* tx1  = ws + 2 * NN + NE;               // byte offset 3.6e6, 16B aligned
  float* t2   = tx1 + (size_t)NN * FD;          // 16B aligned
  const int zero_total = 2 * NN + NE + 2 * NN * FD;  // re-zero everything each call

  zero_kernel<<<2048, 256, 0, stream>>>(ws, zero_total);
  deg_kernel<<<(NE + 255) / 256, 256, 0, stream>>>(ew, src, deg);
  dinv_kernel<<<(NN + 255) / 256, 256, 0, stream>>>(deg, dinv);
  lapw_kernel<<<(NE + 255) / 256, 256, 0, stream>>>(ew, src, dst, dinv, wlap);
  spmm_kernel<<<(NE * 16) / 256, 256, 0, stream>>>(wlap, src, dst, x, tx1);
  spmm_kernel<<<(NE * 16) / 256, 256, 0, stream>>>(wlap, src, dst, tx1, t2);
  dense_kernel<<<(NTILES + 7) / 8, 256, 0, stream>>>(
      x, tx1, t2, Wxz, bxz, bhz, Wxh, bxh, bhh, Wlin, blin, out);
}